// GraphModel_31258771980861
// MI455X (gfx1250) — compile-verified
//
#include <hip/hip_runtime.h>
#include <hip/hip_bf16.h>
#include <math.h>

// ---------------------------------------------------------------------------
// Model constants (from reference)
// ---------------------------------------------------------------------------
#define NN     10000      // nodes
#define NE     320000     // edges
#define BB     16         // batch
#define EEMB   8          // embedding dim
#define CC     32         // hidden channels
// layout: h[n][b][c]  (node-major so each edge moves one contiguous B*C block)

typedef float v2f __attribute__((ext_vector_type(2)));
typedef float v8f __attribute__((ext_vector_type(8)));

// ---------------------------------------------------------------------------
// 1) Embedding: h[n][b][e] = x[b][n] * emb[n][e]
// ---------------------------------------------------------------------------
__global__ void embed_kernel(const float* __restrict__ x,
                             const float* __restrict__ emb,
                             float* __restrict__ h)
{
    int t = blockIdx.x * blockDim.x + threadIdx.x;     // over NN*BB*EEMB
    if (t >= NN * BB * EEMB) return;
    int e  = t & (EEMB - 1);
    int nb = t >> 3;            // n*BB + b
    int b  = nb & (BB - 1);
    int n  = nb >> 4;
    h[t] = x[b * NN + n] * emb[n * EEMB + e];
}

// ---------------------------------------------------------------------------
// 2) Edge scatter-add: agg[dst*RC + 0..RC-1] += h[src*RC + 0..RC-1]
//    (agg pre-initialized to h for the self-loop term via d2d copy)
//    One wave owns 128 contiguous floats of one edge:
//      - edge indices are wave-uniform -> readfirstlane -> scalar s_load
//      - each lane: one b128 load + 4 global_atomic_add_f32
// ---------------------------------------------------------------------------
template <int RC>
__global__ void scatter_kernel(const float* __restrict__ h,
                               const int* __restrict__ esrc,
                               const int* __restrict__ edst,
                               float* __restrict__ agg)
{
    constexpr int WPE = RC / 128;      // waves per edge (1 or 4)
    long long gwave = ((long long)blockIdx.x * blockDim.x + threadIdx.x) >> 5;
    const int lane  = threadIdx.x & 31;
    if (gwave >= (long long)NE * WPE) return;

    int e    = (int)(gwave / WPE);
    int part = (int)(gwave - (long long)e * WPE);
    // wave-uniform by construction; force into SGPRs -> scalar index loads
    e    = __builtin_amdgcn_readfirstlane(e);
    part = __builtin_amdgcn_readfirstlane(part);

    const int s   = esrc[e];
    const int d   = edst[e];
    const int off = part * 128 + lane * 4;

    const float4 v = *(const float4*)(h + (long long)s * RC + off);
    float* dp = agg + (long long)d * RC + off;
    atomicAdd(dp + 0, v.x);
    atomicAdd(dp + 1, v.y);
    atomicAdd(dp + 2, v.z);
    atomicAdd(dp + 3, v.w);
}

// ---------------------------------------------------------------------------
// 3) Per-node linear + gate, fused.  out = relu(g * (agg@W + b)),
//    g = sigmoid((agg@W + b) @ gw + gb).
//    One wave handles a 16-row x 32-col tile via V_WMMA_F32_16X16X4_F32.
//    A 16x4 f32 layout: lanes 0-15 -> M=lane, K pair (kk,kk+1);
//                       lanes 16-31 -> M=lane-16, K pair (kk+2,kk+3).
//    B 4x16 f32 layout: lanes 0-15 -> N=lane, K (kk,kk+1) in v[0],v[1];
//                       lanes 16-31 -> N=lane-16, K (kk+2,kk+3).
//    C/D: vgpr v holds M=v (lanes 0-15) / M=v+8 (lanes 16-31), N=lane&15.
// ---------------------------------------------------------------------------
template <int K>
__global__ void gemm_gate_kernel(const float* __restrict__ agg,  // [N*B][K]
                                 const float* __restrict__ W,    // [K][32]
                                 const float* __restrict__ bias, // [32]
                                 const float* __restrict__ gw,   // [32]
                                 const float* __restrict__ gb,   // [1]
                                 float* __restrict__ out)        // [N*B][32]
{
    __shared__ float lds_tile[4][16 * CC];   // per-wave 16x32 pre-activation
    __shared__ float lds_gate[4][16];        // per-wave row gates

    const int wave = threadIdx.x >> 5;       // 4 waves / block
    const int lane = threadIdx.x & 31;
    const int half = lane >> 4;              // 0: lanes 0-15, 1: lanes 16-31
    const int l    = lane & 15;

    const long long tile = (long long)blockIdx.x * 4 + wave;  // 16-row tile
    const long long row0 = tile * 16;

    v8f acc0 = {};   // cols 0..15
    v8f acc1 = {};   // cols 16..31

    const float* arow = agg + (row0 + l) * K + 2 * half;
#pragma unroll
    for (int kk = 0; kk < K; kk += 4) {
        v2f a = *(const v2f*)(arow + kk);
        int kb = kk + 2 * half;
        v2f b0; b0[0] = W[kb * CC + l];       b0[1] = W[(kb + 1) * CC + l];
        v2f b1; b1[0] = W[kb * CC + 16 + l];  b1[1] = W[(kb + 1) * CC + 16 + l];
        acc0 = __builtin_amdgcn_wmma_f32_16x16x4_f32(
                   false, a, false, b0, (short)0, acc0, false, false);
        acc1 = __builtin_amdgcn_wmma_f32_16x16x4_f32(
                   false, a, false, b1, (short)0, acc1, false, false);
    }

    // bias + stage tile into LDS for the row-wise gate reduction
    const float bias0 = bias[l];
    const float bias1 = bias[16 + l];
#pragma unroll
    for (int v = 0; v < 8; ++v) {
        int M = v + 8 * half;
        lds_tile[wave][M * CC + l]      = acc0[v] + bias0;
        lds_tile[wave][M * CC + 16 + l] = acc1[v] + bias1;
    }
    __syncthreads();

    if (lane < 16) {
        int M = lane;
        float z = gb[0];
#pragma unroll
        for (int j = 0; j < CC; ++j)
            z += lds_tile[wave][M * CC + j] * gw[j];
        lds_gate[wave][M] = 1.0f / (1.0f + __expf(-z));
    }
    __syncthreads();

#pragma unroll
    for (int v = 0; v < 8; ++v) {
        int M = v + 8 * half;
        float g  = lds_gate[wave][M];
        float h0 = g * lds_tile[wave][M * CC + l];
        float h1 = g * lds_tile[wave][M * CC + 16 + l];
        h0 = h0 > 0.0f ? h0 : 0.0f;
        h1 = h1 > 0.0f ? h1 : 0.0f;
        out[(row0 + M) * CC + l]      = h0;
        out[(row0 + M) * CC + 16 + l] = h1;
    }
}

// ---------------------------------------------------------------------------
// 4) Logistic readout: out[b][j] = logb[j] + sum_{n,c} h[n][b][c]*logW[n*32+c][j]
//    Two-stage deterministic reduction (no float atomics on the result).
// ---------------------------------------------------------------------------
#define NCHUNK 40   // ceil(10000/256)

__global__ void logit_partial_kernel(const float* __restrict__ h,     // [N][B][32]
                                     const float* __restrict__ logW,  // [N*32][2]
                                     float* __restrict__ part)        // [B][NCHUNK][2]
{
    __shared__ float r0[256];
    __shared__ float r1[256];
    const int b     = blockIdx.y;
    const int chunk = blockIdx.x;
    const int n     = chunk * 256 + threadIdx.x;

    float s0 = 0.0f, s1 = 0.0f;
    if (n < NN) {
        const float* hr = h + ((long long)n * BB + b) * CC;
        const float* wr = logW + (long long)n * CC * 2;
#pragma unroll
        for (int c = 0; c < CC; ++c) {
            float hv = hr[c];
            s0 += hv * wr[c * 2 + 0];
            s1 += hv * wr[c * 2 + 1];
        }
    }
    r0[threadIdx.x] = s0;
    r1[threadIdx.x] = s1;
    __syncthreads();
    for (int stride = 128; stride > 0; stride >>= 1) {
        if (threadIdx.x < stride) {
            r0[threadIdx.x] += r0[threadIdx.x + stride];
            r1[threadIdx.x] += r1[threadIdx.x + stride];
        }
        __syncthreads();
    }
    if (threadIdx.x == 0) {
        part[(b * NCHUNK + chunk) * 2 + 0] = r0[0];
        part[(b * NCHUNK + chunk) * 2 + 1] = r1[0];
    }
}

__global__ void logit_final_kernel(const float* __restrict__ part,
                                   const float* __restrict__ logb,
                                   float* __restrict__ out)
{
    int t = threadIdx.x;                 // 0..31 -> (b, j)
    if (t >= BB * 2) return;
    int b = t >> 1, j = t & 1;
    float s = logb[j];
    for (int c = 0; c < NCHUNK; ++c)
        s += part[(b * NCHUNK + c) * 2 + j];
    out[b * 2 + j] = s;
}

// ---------------------------------------------------------------------------
// Host-side launch
// ---------------------------------------------------------------------------
extern "C" void kernel_launch(void* const* d_in, const int* in_sizes, int n_in,
                              void* d_out, int out_size, void* d_ws, size_t ws_size,
                              hipStream_t stream)
{
    const float* x    = (const float*)d_in[0];
    const float* emb  = (const float*)d_in[1];
    const float* W0   = (const float*)d_in[2];
    const float* b0   = (const float*)d_in[3];
    const float* gw0  = (const float*)d_in[4];
    const float* gb0  = (const float*)d_in[5];
    const float* W1   = (const float*)d_in[6];
    const float* b1   = (const float*)d_in[7];
    const float* gw1  = (const float*)d_in[8];
    const float* gb1  = (const float*)d_in[9];
    const float* logW = (const float*)d_in[10];
    const float* logb = (const float*)d_in[11];
    const int*   esrc = (const int*)d_in[12];
    const int*   edst = (const int*)d_in[13];
    float*       out  = (float*)d_out;

    const size_t SLOT = (size_t)NN * BB * CC * sizeof(float);   // 20.48 MB
    float* buf0 = (float*)((char*)d_ws + 0 * SLOT);
    float* buf1 = (float*)((char*)d_ws + 1 * SLOT);
    float* buf2 = (float*)((char*)d_ws + 2 * SLOT);
    float* part = (float*)((char*)d_ws + 3 * SLOT);             // 16*40*2 floats

    // --- embed: buf0 = x * emb  [N][B][8] ---
    {
        int total = NN * BB * EEMB;
        embed_kernel<<<(total + 255) / 256, 256, 0, stream>>>(x, emb, buf0);
    }

    // --- layer 0 aggregate: buf1 = buf0 + scatter ---
    hipMemcpyAsync(buf1, buf0, (size_t)NN * BB * EEMB * sizeof(float),
                   hipMemcpyDeviceToDevice, stream);
    {
        // RC=128: 1 wave per edge -> NE*32 threads
        long long total = (long long)NE * 32;
        long long blocks = (total + 255) / 256;
        scatter_kernel<BB * EEMB><<<(unsigned)blocks, 256, 0, stream>>>(
            buf0, esrc, edst, buf1);
    }

    // --- layer 0 linear+gate (WMMA f32, K=8): buf2 = relu(g*(buf1@W0+b0)) ---
    gemm_gate_kernel<EEMB><<<(NN * BB) / (16 * 4), 128, 0, stream>>>(
        buf1, W0, b0, gw0, gb0, buf2);

    // --- layer 1 aggregate: buf0 = buf2 + scatter ---
    hipMemcpyAsync(buf0, buf2, SLOT, hipMemcpyDeviceToDevice, stream);
    {
        // RC=512: 4 waves per edge -> NE*4*32 threads
        long long total = (long long)NE * 4 * 32;
        long long blocks = (total + 255) / 256;
        scatter_kernel<BB * CC><<<(unsigned)blocks, 256, 0, stream>>>(
            buf2, esrc, edst, buf0);
    }

    // --- layer 1 linear+gate (WMMA f32, K=32): buf1 = relu(g*(buf0@W1+b1)) ---
    gemm_gate_kernel<CC><<<(NN * BB) / (16 * 4), 128, 0, stream>>>(
        buf0, W1, b1, gw1, gb1, buf1);

    // --- logistic readout (deterministic two-stage reduction) ---
    {
        dim3 grid(NCHUNK, BB);
        logit_partial_kernel<<<grid, 256, 0, stream>>>(buf1, logW, part);
        logit_final_kernel<<<1, 32, 0, stream>>>(part, logb, out);
    }
}